// ExLoss_71227737637224
// MI455X (gfx1250) — compile-verified
//
#include <hip/hip_runtime.h>
#include <hip/hip_bf16.h>
#include <math.h>

typedef __attribute__((ext_vector_type(2))) float v2f;
typedef __attribute__((ext_vector_type(8))) float v8f;

#define B_ 64
#define F_ 256
#define C_ 100000
#define T_SCALE 1.0f
#define P_MARGIN 0.2f
#define N_MARGIN 0.3f
#define NTILES (C_ / 16)   // 6250 column tiles of 16

// ---------------------------------------------------------------------------
// Kernel 1: logits[64 x 100000] = inputs[64x256] @ V[100000x256]^T * T
// One wave per 16-column tile, all 64 rows (4 accumulator tiles) so V streams
// through HBM exactly once. fp32 WMMA 16x16x4 (exact fp32 math).
//
// A (16x4 f32, 2 VGPR): lane L holds M = L&15; VGPR0 = K = 2*(L>>4),
//                       VGPR1 = K = 2*(L>>4)+1.
// B (4x16 f32, 2 VGPR): lane L holds N = L&15; same K split.
// D (16x16 f32, 8 VGPR): VGPR r of lane L = element (M = r + 8*(L>>4), N = L&15).
// ---------------------------------------------------------------------------
__global__ __launch_bounds__(256) void exloss_gemm(
    const float* __restrict__ inputs,   // [64 x 256] row-major
    const float* __restrict__ V,        // [100000 x 256] row-major
    float* __restrict__ logits)         // [64 x 100000] row-major
{
    const int lane = threadIdx.x & 31;
    const int wave = threadIdx.x >> 5;
    const int tile = blockIdx.x * 8 + wave;
    if (tile >= NTILES) return;                 // wave-uniform exit: EXEC stays all-1s

    const int n = lane & 15;                    // M for A, N for B
    const int h = lane >> 4;                    // K-half select

    const float* __restrict__ brow = V + (long)(tile * 16 + n) * F_ + 2 * h;
    const float* __restrict__ arow = inputs + (long)n * F_ + 2 * h;

    v8f acc[4];
    acc[0] = (v8f){}; acc[1] = (v8f){}; acc[2] = (v8f){}; acc[3] = (v8f){};

#pragma unroll 4
    for (int k = 0; k < F_; k += 4) {
        v2f b  = *(const v2f*)(brow + k);
        v2f a0 = *(const v2f*)(arow + k);
        v2f a1 = *(const v2f*)(arow + 16 * F_ + k);
        v2f a2 = *(const v2f*)(arow + 32 * F_ + k);
        v2f a3 = *(const v2f*)(arow + 48 * F_ + k);
        acc[0] = __builtin_amdgcn_wmma_f32_16x16x4_f32(false, a0, false, b, (short)0, acc[0], false, false);
        acc[1] = __builtin_amdgcn_wmma_f32_16x16x4_f32(false, a1, false, b, (short)0, acc[1], false, false);
        acc[2] = __builtin_amdgcn_wmma_f32_16x16x4_f32(false, a2, false, b, (short)0, acc[2], false, false);
        acc[3] = __builtin_amdgcn_wmma_f32_16x16x4_f32(false, a3, false, b, (short)0, acc[3], false, false);
    }

    // Store D tiles: row = rt*16 + 8*h + r, col = tile*16 + n (coalesced over n)
    const long colbase = (long)tile * 16 + n;
#pragma unroll
    for (int rt = 0; rt < 4; ++rt) {
#pragma unroll
        for (int r = 0; r < 8; ++r) {
            long row = rt * 16 + 8 * h + r;
            logits[row * (long)C_ + colbase] = acc[rt][r] * T_SCALE;
        }
    }
}

// ---------------------------------------------------------------------------
// Kernel 2: per-row  lse(b) - logits[b, targets[b]]  -> ws[b]   (b = 0..63)
// Deterministic LDS tree reductions, two-pass (max, then sum-exp).
// ---------------------------------------------------------------------------
__global__ __launch_bounds__(256) void exloss_row_nll(
    const float* __restrict__ logits,
    const int* __restrict__ targets,
    float* __restrict__ ws)
{
    __shared__ float red[256];
    const int b = blockIdx.x;
    const int t = threadIdx.x;
    const float* __restrict__ row = logits + (long)b * C_;

    float m = -INFINITY;
    for (int c = t; c < C_; c += 256) m = fmaxf(m, row[c]);
    red[t] = m;
    __syncthreads();
    for (int s = 128; s > 0; s >>= 1) {
        if (t < s) red[t] = fmaxf(red[t], red[t + s]);
        __syncthreads();
    }
    const float rowmax = red[0];
    __syncthreads();

    float sum = 0.0f;
    for (int c = t; c < C_; c += 256) sum += expf(row[c] - rowmax);
    red[t] = sum;
    __syncthreads();
    for (int s = 128; s > 0; s >>= 1) {
        if (t < s) red[t] += red[t + s];
        __syncthreads();
    }
    if (t == 0) {
        float lse = rowmax + logf(red[0]);
        ws[b] = lse - row[targets[b]];
    }
}

// ---------------------------------------------------------------------------
// Kernel 3: pairwise cosine sims + hard-mining BCE losses -> ws[64]
// Single 1024-thread block; everything in LDS; fixed-order reductions.
// ---------------------------------------------------------------------------
__global__ __launch_bounds__(1024) void exloss_pairwise(
    const float* __restrict__ inputs,
    const unsigned char* __restrict__ pos_mask,   // jax bool: 1 byte/elem
    const unsigned char* __restrict__ neg_mask,
    float* __restrict__ ws)
{
    __shared__ float inv_norm[B_];
    __shared__ float sims[B_ * B_];
    __shared__ float p_thr[B_], n_thr[B_];
    __shared__ float r_hp_s[1024], r_hn_s[1024];
    __shared__ int   r_hp_c[1024], r_hn_c[1024];

    const int t = threadIdx.x;

    if (t < B_) {
        const float* r = inputs + t * F_;
        float s = 0.0f;
        for (int k = 0; k < F_; ++k) s += r[k] * r[k];
        inv_norm[t] = 1.0f / sqrtf(s);
    }
    __syncthreads();

    for (int e = t; e < B_ * B_; e += 1024) {
        const int i = e >> 6, j = e & 63;
        const float* ri = inputs + i * F_;
        const float* rj = inputs + j * F_;
        float d = 0.0f;
        for (int k = 0; k < F_; ++k) d += ri[k] * rj[k];
        d *= inv_norm[i] * inv_norm[j];
        sims[e] = fminf(fmaxf(d, -1.0f), 1.0f);
    }
    __syncthreads();

    if (t < B_) {
        float nmin = INFINITY, pmax = -INFINITY;
        for (int j = 0; j < B_; ++j) {
            if (j == t) continue;
            const float s  = sims[t * B_ + j];
            const float ps = pos_mask[t * B_ + j] ? s : 2.0f;       // sentinel
            nmin = fminf(nmin, ps);                                  // n_thrds uses psims
            const float thd = (ps == 2.0f) ? -2.0f : ps;             // flip sentinel
            pmax = fmaxf(pmax, thd);
        }
        n_thr[t] = nmin - N_MARGIN;
        p_thr[t] = pmax - P_MARGIN;
    }
    __syncthreads();

    // Sentinel 2.0 can never satisfy "< p_thr (<=0.8)" or "< n_thr (<=1.7)",
    // so masks reduce to mask && offdiag && sims < thr; vals are the real sims.
    float hp_s = 0.0f, hn_s = 0.0f;
    int hp_c = 0, hn_c = 0;
    for (int e = t; e < B_ * B_; e += 1024) {
        const int i = e >> 6, j = e & 63;
        if (i == j) continue;
        const float s = sims[e];
        if (pos_mask[e] && s < p_thr[i]) { hp_s += log1pf(expf(-s)); ++hp_c; }
        if (neg_mask[e] && s < n_thr[i]) { hn_s += log1pf(expf(-s)); ++hn_c; }
    }
    r_hp_s[t] = hp_s; r_hn_s[t] = hn_s; r_hp_c[t] = hp_c; r_hn_c[t] = hn_c;
    __syncthreads();
    for (int s = 512; s > 0; s >>= 1) {
        if (t < s) {
            r_hp_s[t] += r_hp_s[t + s];
            r_hn_s[t] += r_hn_s[t + s];
            r_hp_c[t] += r_hp_c[t + s];
            r_hn_c[t] += r_hn_c[t + s];
        }
        __syncthreads();
    }
    if (t == 0) {
        float hp = (r_hp_c[0] > 0) ? r_hp_s[0] / (float)r_hp_c[0] : 0.0f;
        float hn = (r_hn_c[0] > 0) ? r_hn_s[0] / (float)r_hn_c[0] : 0.0f;
        ws[B_] = hp + hn;
    }
}

// ---------------------------------------------------------------------------
// Kernel 4: loss = mean(ws[0..63]) + ws[64] -> out[0]  (fixed-order sum)
// ---------------------------------------------------------------------------
__global__ void exloss_finalize(const float* __restrict__ ws, float* __restrict__ out)
{
    if (threadIdx.x == 0 && blockIdx.x == 0) {
        float s = 0.0f;
        for (int b = 0; b < B_; ++b) s += ws[b];
        out[0] = s / (float)B_ + ws[B_];
    }
}

extern "C" void kernel_launch(void* const* d_in, const int* in_sizes, int n_in,
                              void* d_out, int out_size, void* d_ws, size_t ws_size,
                              hipStream_t stream) {
    const float*         inputs   = (const float*)d_in[0];         // [64,256] f32
    const int*           targets  = (const int*)d_in[1];           // [64] i32
    const unsigned char* pos_mask = (const unsigned char*)d_in[2]; // [64,64] bool
    const unsigned char* neg_mask = (const unsigned char*)d_in[3]; // [64,64] bool
    const float*         V        = (const float*)d_in[4];         // [100000,256] f32

    float* out    = (float*)d_out;
    float* logits = out + 1;          // tuple return: (loss, logits) flat-concat
    float* ws     = (float*)d_ws;     // [0..63] row NLL partials, [64] hp+hn

    // 6250 tiles / 8 waves per 256-thread block
    exloss_gemm    <<<(NTILES + 7) / 8, 256, 0, stream>>>(inputs, V, logits);
    exloss_row_nll <<<B_,               256, 0, stream>>>(logits, targets, ws);
    exloss_pairwise<<<1,               1024, 0, stream>>>(inputs, pos_mask, neg_mask, ws);
    exloss_finalize<<<1,                 64, 0, stream>>>(ws, out);
}